// LCOnlyModel_33681133536025
// MI455X (gfx1250) — compile-verified
//
#include <hip/hip_runtime.h>

typedef __attribute__((ext_vector_type(2))) float v2f;
typedef __attribute__((ext_vector_type(8))) float v8f;

#define NVEH 8192
#define NT   512
#define OFF_LC  (NVEH * NT)                 //  4,194,304
#define OFF_H   (OFF_LC + NVEH * NT * 3)    // 16,777,216
#define OFF_C   (OFF_H + NVEH * 20)         // 16,941,056

__device__ __forceinline__ float sigm_f(float x) {
    return __builtin_amdgcn_rcpf(1.0f + __expf(-x));
}
__device__ __forceinline__ float tanh_f(float x) {
    return 1.0f - 2.0f * __builtin_amdgcn_rcpf(__expf(2.0f * x) + 1.0f);
}

// One wave (32 lanes) per block; wave handles 16 vehicles across all 512 steps.
__global__ __launch_bounds__(32) void lstm_lc_kernel(
    const float* __restrict__ lf_in,   // (NVEH, NT, 12)
    const float* __restrict__ traj,    // (NVEH, NT)
    const float* __restrict__ hs,      // (2, NVEH, 20)
    const float* __restrict__ Wk,      // (12, 80)
    const float* __restrict__ Wr,      // (20, 80)
    const float* __restrict__ bl,      // (80)
    const float* __restrict__ Wd,      // (20, 10)
    const float* __restrict__ bd_,     // (10)
    const float* __restrict__ Wlc,     // (10, 3)
    const float* __restrict__ blc_,    // (3)
    float* __restrict__ out)
{
    const int lane = threadIdx.x;
    const int half = lane >> 4;        // 0: lanes 0-15, 1: lanes 16-31
    const int l16  = lane & 15;        // vehicle-in-wave for A/D rows, column for B/D
    const int vg   = blockIdx.x * 16 + l16;   // global vehicle id
    const int u0   = 10 * half;        // unit range handled by this lane (gates phase)

    // LDS staging (strides chosen: 16B alignment + distinct banks across 16 rows)
    __shared__ float xh[16 * 36];      // per-vehicle [x(12) | h(20)] row, stride 36
    __shared__ float zb[16 * 81];      // z pre-activations (80 cols), stride 81
    __shared__ float db[16 * 14];      // relu dense (10 cols + 2 zero pad), stride 14
    __shared__ float lcb[16 * 52];     // lc accumulation, 48 floats/veh per 16 steps

    // ---- Preload all weights into registers (B-operand layout: 4x16 tile,
    //      VGPR0 = rows K,K+2 split by lane half, VGPR1 = rows K+1,K+3) ----
    v2f Bz[8][5];
    #pragma unroll
    for (int kk = 0; kk < 8; ++kk) {
        const int r0 = 4 * kk + 2 * half;
        const int r1 = r0 + 1;
        #pragma unroll
        for (int tt = 0; tt < 5; ++tt) {
            const int col = 16 * tt + l16;
            float w0 = (r0 < 12) ? Wk[r0 * 80 + col] : Wr[(r0 - 12) * 80 + col];
            float w1 = (r1 < 12) ? Wk[r1 * 80 + col] : Wr[(r1 - 12) * 80 + col];
            Bz[kk][tt] = (v2f){w0, w1};
        }
    }
    v2f Bd[5];
    #pragma unroll
    for (int kk = 0; kk < 5; ++kk) {
        const int r0 = 4 * kk + 2 * half;
        float w0 = (l16 < 10) ? Wd[r0 * 10 + l16] : 0.0f;
        float w1 = (l16 < 10) ? Wd[(r0 + 1) * 10 + l16] : 0.0f;
        Bd[kk] = (v2f){w0, w1};
    }
    v2f Blc[3];
    #pragma unroll
    for (int kk = 0; kk < 3; ++kk) {
        const int r0 = 4 * kk + 2 * half;
        float w0 = (r0 < 10 && l16 < 3) ? Wlc[r0 * 3 + l16] : 0.0f;
        float w1 = (r0 + 1 < 10 && l16 < 3) ? Wlc[(r0 + 1) * 3 + l16] : 0.0f;
        Blc[kk] = (v2f){w0, w1};
    }
    float bz[5];
    #pragma unroll
    for (int tt = 0; tt < 5; ++tt) bz[tt] = bl[16 * tt + l16];
    const float bD = (l16 < 10) ? bd_[l16] : 0.0f;
    const float bL = (l16 < 3) ? blc_[l16] : 0.0f;

    // ---- Initial state ----
    float cst[10];
    #pragma unroll
    for (int j = 0; j < 10; ++j)
        cst[j] = hs[(size_t)NVEH * 20 + (size_t)vg * 20 + u0 + j];
    #pragma unroll
    for (int j = 0; j < 10; ++j)
        xh[l16 * 36 + 12 + u0 + j] = hs[(size_t)vg * 20 + u0 + j];
    __syncthreads();

    const float* inv = lf_in + (size_t)vg * NT * 12;

    for (int t = 0; t < NT; ++t) {
        // ---- Phase 1: build x (lower half lanes; 1 vehicle each) ----
        if (lane < 16) {
            float pos = traj[(size_t)vg * NT + t];
            const float4* p4 = reinterpret_cast<const float4*>(inv + (size_t)t * 12);
            float4 a = p4[0], b = p4[1], c2 = p4[2];
            float x[12];
            const float iHD = 1.0f / 100.0f, iV = 1.0f / 40.0f;
            x[0] = (a.x - pos) * iHD;  x[1] = (a.y - pos) * iHD;  x[2] = (a.z - pos) * iHD;
            x[3] = (pos - a.w) * iHD;  x[4] = (pos - b.x) * iHD;  x[5] = (pos - b.y) * iHD;
            x[6] = b.z * iV;  x[7] = b.w * iV;
            x[8] = c2.x * iV; x[9] = c2.y * iV; x[10] = c2.z * iV; x[11] = c2.w * iV;
            #pragma unroll
            for (int j = 0; j < 12; ++j) if (x[j] != x[j]) x[j] = 1.0f;
            float4* xr = reinterpret_cast<float4*>(&xh[l16 * 36]);
            xr[0] = make_float4(x[0], x[1], x[2], x[3]);
            xr[1] = make_float4(x[4], x[5], x[6], x[7]);
            xr[2] = make_float4(x[8], x[9], x[10], x[11]);
        }
        __syncthreads();

        // ---- Phase 2: z = [x|h] @ [Wk;Wr]  (16x80, K=32) via WMMA f32 16x16x4 ----
        v8f acc[5] = {};
        #pragma unroll
        for (int kk = 0; kk < 8; ++kk) {
            v2f a = *reinterpret_cast<const v2f*>(&xh[l16 * 36 + 4 * kk + 2 * half]);
            #pragma unroll
            for (int tt = 0; tt < 5; ++tt) {
                acc[tt] = __builtin_amdgcn_wmma_f32_16x16x4_f32(
                    false, a, false, Bz[kk][tt], (short)0, acc[tt], false, false);
            }
        }
        // D layout: VGPR r, lane l -> M = r + 8*half, N = l&15
        #pragma unroll
        for (int tt = 0; tt < 5; ++tt) {
            #pragma unroll
            for (int r = 0; r < 8; ++r)
                zb[(r + 8 * half) * 81 + 16 * tt + l16] = acc[tt][r] + bz[tt];
        }
        __syncthreads();

        // ---- Phase 3: gate nonlinearities; c in regs; h -> xh[12..31] ----
        #pragma unroll
        for (int j = 0; j < 10; ++j) {
            const int base = l16 * 81;
            const int u = u0 + j;
            float zi = zb[base + u];
            float zf = zb[base + 20 + u];
            float zg = zb[base + 40 + u];
            float zo = zb[base + 60 + u];
            float cn = sigm_f(zf) * cst[j] + sigm_f(zi) * tanh_f(zg);
            cst[j] = cn;
            xh[l16 * 36 + 12 + u] = sigm_f(zo) * tanh_f(cn);
        }
        __syncthreads();

        // ---- Phase 4: d = relu(h @ Wd + bd)  (K=20 -> 5 wmma) ----
        v8f ad = {};
        #pragma unroll
        for (int kk = 0; kk < 5; ++kk) {
            v2f a = *reinterpret_cast<const v2f*>(&xh[l16 * 36 + 12 + 4 * kk + 2 * half]);
            ad = __builtin_amdgcn_wmma_f32_16x16x4_f32(
                false, a, false, Bd[kk], (short)0, ad, false, false);
        }
        if (l16 < 10) {
            #pragma unroll
            for (int r = 0; r < 8; ++r)
                db[(r + 8 * half) * 14 + l16] = fmaxf(ad[r] + bD, 0.0f);
        } else if (l16 < 12) {
            #pragma unroll
            for (int r = 0; r < 8; ++r)
                db[(r + 8 * half) * 14 + l16] = 0.0f;   // K-pad rows 10,11
        }
        __syncthreads();

        // ---- Phase 5: lc = d @ Wlc + blc  (K=12 padded -> 3 wmma) ----
        v8f al = {};
        #pragma unroll
        for (int kk = 0; kk < 3; ++kk) {
            v2f a = *reinterpret_cast<const v2f*>(&db[l16 * 14 + 4 * kk + 2 * half]);
            al = __builtin_amdgcn_wmma_f32_16x16x4_f32(
                false, a, false, Blc[kk], (short)0, al, false, false);
        }
        if (l16 < 3) {
            #pragma unroll
            for (int r = 0; r < 8; ++r)
                lcb[(r + 8 * half) * 52 + (t & 15) * 3 + l16] = al[r] + bL;
        }
        __syncthreads();

        // ---- Phase 6: coalesced flush of 16 steps of lc per vehicle ----
        if ((t & 15) == 15) {
            float* gb = out + OFF_LC + (size_t)vg * (NT * 3) + (size_t)(t - 15) * 3;
            #pragma unroll
            for (int j = 0; j < 6; ++j) {
                const int q = half + 2 * j;    // 12 float4 chunks split across halves
                float4 v = *reinterpret_cast<const float4*>(&lcb[l16 * 52 + q * 4]);
                *reinterpret_cast<float4*>(gb + q * 4) = v;
            }
            __syncthreads();
        }
    }

    // ---- Final h, c ----
    #pragma unroll
    for (int j = 0; j < 10; ++j)
        out[OFF_C + (size_t)vg * 20 + u0 + j] = cst[j];
    #pragma unroll
    for (int j = 0; j < 10; ++j)
        out[OFF_H + (size_t)vg * 20 + u0 + j] = xh[l16 * 36 + 12 + u0 + j];
}

__global__ void copy_traj_kernel(const float4* __restrict__ src,
                                 float4* __restrict__ dst, int n4)
{
    int i = blockIdx.x * blockDim.x + threadIdx.x;
    if (i < n4) dst[i] = src[i];
}

extern "C" void kernel_launch(void* const* d_in, const int* in_sizes, int n_in,
                              void* d_out, int out_size, void* d_ws, size_t ws_size,
                              hipStream_t stream) {
    const float* lf_in = (const float*)d_in[0];
    const float* traj  = (const float*)d_in[1];
    const float* hs    = (const float*)d_in[2];
    const float* Wk    = (const float*)d_in[3];
    const float* Wr    = (const float*)d_in[4];
    const float* bl    = (const float*)d_in[5];
    const float* Wd    = (const float*)d_in[6];
    const float* bd    = (const float*)d_in[7];
    const float* Wlc   = (const float*)d_in[8];
    const float* blc   = (const float*)d_in[9];
    float* out = (float*)d_out;

    // Output 0: straight copy of true_traj (NVEH*NT floats)
    const int n4 = (NVEH * NT) / 4;
    copy_traj_kernel<<<(n4 + 255) / 256, 256, 0, stream>>>(
        (const float4*)traj, (float4*)out, n4);

    // Main fused LSTM + dense heads: 512 blocks x 32 threads (1 wave, 16 vehicles)
    lstm_lc_kernel<<<NVEH / 16, 32, 0, stream>>>(
        lf_in, traj, hs, Wk, Wr, bl, Wd, bd, Wlc, blc, out);
}